// ArmaCell_73693048865344
// MI455X (gfx1250) — compile-verified
//
#include <hip/hip_runtime.h>
#include <stdint.h>

// ---------------------------------------------------------------------------
// ARMA cell on gfx1250: V_WMMA_F32_16X16X4_F32 + async global->LDS staging
// with double buffering (GLOBAL_LOAD_ASYNC_TO_LDS_B32, ASYNCcnt).
// W tile stored TRANSPOSED in LDS so B fragments are single ds_load_b64
// pairs (no register-shuffle fixups feeding the WMMAs).
//   inputs : (B=1024, K=256, P=4)          f32
//   state  : (B, K*UNITS=8192, Q=4)        f32
//   kernel : (P=4, UNITS=32, K=256, K=256) f32
//   rkernel: (Q=4, UNITS=32, K=256, K=256) f32
//   out       = (B, 8192)    : batched-over-u GEMM (1024x2048)@(2048x256)
//   out_state = (B, 8192, 4) : [out, state[:,:,0:3]]
// ---------------------------------------------------------------------------

typedef float v2f __attribute__((ext_vector_type(2)));
typedef float v8f __attribute__((ext_vector_type(8)));

#define B_SZ   1024
#define KDIM   256
#define UNITS  32
#define PDIM   4
#define QDIM   4
#define MOUT   (KDIM * UNITS)        // 8192

#define TM 128                       // b-rows per workgroup
#define TN 64                        // j-cols per workgroup
#define TK 16                        // k slice staged in LDS per step
#define A_STRIDE 18                  // padded: conflict-free b64 fragment loads
#define WT_STRIDE 18                 // transposed W tile: WT[col][k], same geometry
#define NSTEPS 128                   // 2048 / TK

// --- CDNA5 async global->LDS copies (ASYNCcnt-tracked, bypass VGPRs) -------
__device__ __forceinline__ void async_b32(uint32_t lds_off, uint64_t gaddr) {
    asm volatile("global_load_async_to_lds_b32 %0, %1, off"
                 :: "v"(lds_off), "v"(gaddr) : "memory");
}
__device__ __forceinline__ void wait_async_le12(void) {
    asm volatile("s_wait_asynccnt 0xC" ::: "memory");
}
__device__ __forceinline__ void wait_async_0(void) {
    asm volatile("s_wait_asynccnt 0x0" ::: "memory");
}

__global__ __launch_bounds__(256)
void arma_wmma_async_kernel(const float* __restrict__ inputs,
                            const float* __restrict__ state,
                            const float* __restrict__ kernelw,
                            const float* __restrict__ rkernel,
                            float* __restrict__ out,
                            float* __restrict__ out_state)
{
    __shared__ float Asm[2][TM * A_STRIDE];    // A tile (b x k-slice), double-buffered
    __shared__ float Wsm[2][TN * WT_STRIDE];   // W tile TRANSPOSED (j x k-slice)

    const int u   = blockIdx.z;                // unit
    const int j0  = blockIdx.x * TN;           // j tile origin
    const int b0  = blockIdx.y * TM;           // batch tile origin

    const int tid  = threadIdx.x;
    const int wave = tid >> 5;                 // 0..7
    const int lane = tid & 31;
    const int half = lane >> 4;
    const int ln   = lane & 15;

    const int r0 = tid >> 4;                   // 0..15 : A row base / W k-row
    const int c0 = tid & 15;                   // A col  / W col4 base

    const uint32_t aOff[2] = { (uint32_t)(uintptr_t)&Asm[0][0],
                               (uint32_t)(uintptr_t)&Asm[1][0] };
    const uint32_t wOff[2] = { (uint32_t)(uintptr_t)&Wsm[0][0],
                               (uint32_t)(uintptr_t)&Wsm[1][0] };

    // Per-thread invariant global bases
    const float* aInBase = inputs + (size_t)(b0 + r0) * (KDIM * PDIM) + c0 * PDIM;
    const float* aStBase = state  + (size_t)(b0 + r0) * (MOUT * QDIM) + (size_t)c0 * (UNITS * QDIM)
                                  + (size_t)u * QDIM;

    // Stage s covers reduction t in [s*TK, s*TK+TK): c = s>>4, ib = (s&15)*TK.
    auto issue_stage = [&](int s, int buf) {
        const int c  = s >> 4;
        const int ib = (s & 15) * TK;
        const uint32_t aO = aOff[buf];
        const uint32_t wO = wOff[buf];
        // ---- A tile: 8 per-lane b32 async copies (strided gather)
        if (c < 4) {
            const float* g = aInBase + (size_t)ib * PDIM + c;
#pragma unroll
            for (int it = 0; it < 8; ++it)
                async_b32(aO + (uint32_t)(((r0 + 16 * it) * A_STRIDE + c0) * 4),
                          (uint64_t)(uintptr_t)(g + (size_t)it * 16 * (KDIM * PDIM)));
        } else {
            const float* g = aStBase + (size_t)ib * (UNITS * QDIM) + (c - 4);
#pragma unroll
            for (int it = 0; it < 8; ++it)
                async_b32(aO + (uint32_t)(((r0 + 16 * it) * A_STRIDE + c0) * 4),
                          (uint64_t)(uintptr_t)(g + (size_t)it * 16 * (MOUT * QDIM)));
        }
        // ---- W tile, transposed store: thread (r0,c0) reads global row (ib+r0),
        //      cols j0 + c0*4 + d, writes WT[c0*4+d][r0]  (4 b32 asyncs)
        const float* wb = (c < 4)
            ? kernelw + (size_t)c       * (UNITS * KDIM * KDIM) + (size_t)u * (KDIM * KDIM)
            : rkernel + (size_t)(c - 4) * (UNITS * KDIM * KDIM) + (size_t)u * (KDIM * KDIM);
        const float* gw = wb + (size_t)(ib + r0) * KDIM + j0 + c0 * 4;
#pragma unroll
        for (int d = 0; d < 4; ++d)
            async_b32(wO + (uint32_t)((((c0 * 4 + d) * WT_STRIDE) + r0) * 4),
                      (uint64_t)(uintptr_t)(gw + d));
    };

    v8f acc[4];
#pragma unroll
    for (int nt = 0; nt < 4; ++nt)
        acc[nt] = (v8f){0.f, 0.f, 0.f, 0.f, 0.f, 0.f, 0.f, 0.f};

    issue_stage(0, 0);                         // prologue: fill buffer 0

    for (int s = 0; s < NSTEPS; ++s) {
        const int cur = s & 1;
        if (s + 1 < NSTEPS) {
            issue_stage(s + 1, cur ^ 1);       // prefetch next stage into other buffer
            wait_async_le12();                 // 12 outstanding = next stage only
        } else {
            wait_async_0();
        }
        __syncthreads();                       // cur buffer visible to all waves

        const float* At = &Asm[cur][0];
        const float* Wt = &Wsm[cur][0];
#pragma unroll
        for (int ks = 0; ks < 4; ++ks) {
            const int k0   = ks * 4 + half * 2;
            const int arow = wave * 16 + ln;
            v2f a = *(const v2f*)(At + arow * A_STRIDE + k0);        // ds_load_b64
#pragma unroll
            for (int nt = 0; nt < 4; ++nt) {
                v2f bf = *(const v2f*)(Wt + (nt * 16 + ln) * WT_STRIDE + k0); // ds_load_b64
                acc[nt] = __builtin_amdgcn_wmma_f32_16x16x4_f32(
                    /*neg_a=*/false, a, /*neg_b=*/false, bf,
                    /*c_mod=*/(short)0, acc[nt],
                    /*reuse_a=*/false, /*reuse_b=*/false);
            }
        }
        __syncthreads();                       // all waves done reading cur buffer
    }

    // ---- epilogue: D layout -> (b, m); fuse the state shift (aligned float4 store)
#pragma unroll
    for (int nt = 0; nt < 4; ++nt) {
        const int j = j0 + nt * 16 + ln;
        const int m = j * UNITS + u;           // out index, k-major (j,u)
#pragma unroll
        for (int r = 0; r < 8; ++r) {
            const int   b  = b0 + wave * 16 + half * 8 + r;
            const float v  = acc[nt][r];
            const size_t so = (size_t)b * (MOUT * QDIM) + (size_t)m * QDIM;

            out[(size_t)b * MOUT + m] = v;

            const float4 s4 = *(const float4*)(state + so);   // state[b][m][0..3]
            float4 o4;
            o4.x = v;                                          // out_state[...,0] = out
            o4.y = s4.x;                                       // shift: state[...,0..2]
            o4.z = s4.y;
            o4.w = s4.z;
            *(float4*)(out_state + so) = o4;
        }
    }
}

extern "C" void kernel_launch(void* const* d_in, const int* in_sizes, int n_in,
                              void* d_out, int out_size, void* d_ws, size_t ws_size,
                              hipStream_t stream) {
    (void)in_sizes; (void)n_in; (void)d_ws; (void)ws_size; (void)out_size;

    const float* inputs  = (const float*)d_in[0];
    const float* state   = (const float*)d_in[1];
    const float* kernelw = (const float*)d_in[2];
    const float* rkernel = (const float*)d_in[3];

    float* out       = (float*)d_out;                 // (B, 8192)
    float* out_state = out + (size_t)B_SZ * MOUT;     // (B, 8192, 4)

    dim3 grid(KDIM / TN, B_SZ / TM, UNITS);           // (4, 8, 32)
    dim3 block(256);
    arma_wmma_async_kernel<<<grid, block, 0, stream>>>(inputs, state, kernelw, rkernel,
                                                       out, out_state);
}